// DenseSparsePreEmbedding_52621939310811
// MI455X (gfx1250) — compile-verified
//
#include <hip/hip_runtime.h>

typedef float v2f __attribute__((ext_vector_type(2)));
typedef float v4f __attribute__((ext_vector_type(4)));
typedef float v8f __attribute__((ext_vector_type(8)));

#define E_DIM 128
#define KDIM  128

// ---------------- slot map construction ----------------
__global__ void init_slot_kernel(int* __restrict__ slot, int n) {
    int i = blockIdx.x * blockDim.x + threadIdx.x;
    if (i < n) slot[i] = -1;
}

// packed = (key<<6)|val. atomicMax => highest key wins (matches sequential
// .at[idx].set over keys); within-key duplicates resolve deterministically.
__global__ void scatter_slot_kernel(const int* __restrict__ i0, const int* __restrict__ v0,
                                    const int* __restrict__ i1, const int* __restrict__ v1,
                                    const int* __restrict__ i2, const int* __restrict__ v2,
                                    const int* __restrict__ i3, const int* __restrict__ v3,
                                    int* __restrict__ slot, int m) {
    int t = blockIdx.x * blockDim.x + threadIdx.x;
    if (t >= 4 * m) return;
    int k = t / m, e = t - k * m;
    const int* idx; const int* val;
    switch (k) {
        case 0: idx = i0; val = v0; break;
        case 1: idx = i1; val = v1; break;
        case 2: idx = i2; val = v2; break;
        default: idx = i3; val = v3; break;
    }
    int packed = (k << 6) | (val[e] & 63);
    atomicMax(&slot[idx[e]], packed);
}

// ---------------- fp32 WMMA GEMM:  C[M,128] = A[M,128] @ B[128,128] (+bias) ----
// One wave computes one 16x16 tile of C via 32 chained v_wmma_f32_16x16x4_f32.
__global__ void wmma_gemm_f32_kernel(const float* __restrict__ A,
                                     const float* __restrict__ B,
                                     const float* __restrict__ bias,
                                     float* __restrict__ C,
                                     int Mrows) {
    const int lane = threadIdx.x & 31;
    const int wave = threadIdx.x >> 5;
    const int nTilesN = E_DIM / 16;                 // 8
    const int tile = blockIdx.x * (blockDim.x >> 5) + wave;
    const int totalTiles = (Mrows / 16) * nTilesN;
    if (tile >= totalTiles) return;
    const int mt = tile / nTilesN;
    const int nt = tile - mt * nTilesN;

    const int m    = lane & 15;                     // row within A tile / col within B tile
    const int half = lane >> 4;                     // 0: K={0,1}, 1: K={2,3}
    const int kOff = half * 2;

    const float* Arow = A + (size_t)(mt * 16 + m) * KDIM;
    const int    ncol = nt * 16 + m;

    v8f acc = {};
    #pragma unroll 4
    for (int k0 = 0; k0 < KDIM; k0 += 4) {
        v2f a, bm;
        a.x  = Arow[k0 + kOff + 0];
        a.y  = Arow[k0 + kOff + 1];
        bm.x = B[(size_t)(k0 + kOff + 0) * E_DIM + ncol];
        bm.y = B[(size_t)(k0 + kOff + 1) * E_DIM + ncol];
        acc = __builtin_amdgcn_wmma_f32_16x16x4_f32(
            /*neg_a=*/false, a, /*neg_b=*/false, bm,
            /*c_mod=*/(short)0, acc, /*reuse_a=*/false, /*reuse_b=*/false);
    }

    const float bv = bias ? bias[ncol] : 0.0f;
    #pragma unroll
    for (int j = 0; j < 8; ++j) {
        const int row = mt * 16 + j + half * 8;     // C/D layout: lanes16-31 hold M=j+8
        C[(size_t)row * E_DIM + ncol] = acc[j] + bv;
    }
}

// ---------------- hot kernel: per-row gather-add (L2-resident tables) ------
// One wave per token row; lane l handles float4 chunk l (32*4 = 128 floats).
__global__ void gather_add_kernel(const int* __restrict__ ff,
                                  const int* __restrict__ slot,
                                  const v4f* __restrict__ Q,
                                  const v4f* __restrict__ P,
                                  v4f* __restrict__ out, int n) {
    const int lane = threadIdx.x & 31;
    const int row  = blockIdx.x * (blockDim.x >> 5) + (threadIdx.x >> 5);
    if (row >= n) return;
    const int f = ff[row];
    const int s = slot[row];                        // wave-uniform branch below
    v4f r = Q[(size_t)f * 32 + lane];
    if (s >= 0) {
        const v4f p = P[(size_t)s * 32 + lane];
        r += p;
    }
    __builtin_nontemporal_store(r, &out[(size_t)row * 32 + lane]);
}

// ---------------- launch ----------------
extern "C" void kernel_launch(void* const* d_in, const int* in_sizes, int n_in,
                              void* d_out, int out_size, void* d_ws, size_t ws_size,
                              hipStream_t stream) {
    // Input order per setup_inputs():
    // 0 fixed_features, 1..8 (idx0,val0,...,idx3,val3),
    // 9 fixed_table, 10..13 tab0..tab3, 14 W_fixed, 15 W_sparse, 16 b
    const int*   ff      = (const int*)d_in[0];
    const int*   idx[4]  = {(const int*)d_in[1], (const int*)d_in[3],
                            (const int*)d_in[5], (const int*)d_in[7]};
    const int*   val[4]  = {(const int*)d_in[2], (const int*)d_in[4],
                            (const int*)d_in[6], (const int*)d_in[8]};
    const float* ftab    = (const float*)d_in[9];
    const float* tabs[4] = {(const float*)d_in[10], (const float*)d_in[11],
                            (const float*)d_in[12], (const float*)d_in[13]};
    const float* Wf      = (const float*)d_in[14];
    const float* Ws      = (const float*)d_in[15];
    const float* bias    = (const float*)d_in[16];

    const int N = in_sizes[0];          // 1,000,000
    const int M = in_sizes[1];          // 125,000
    const int V = in_sizes[9] / KDIM;   // 2048
    const int C = in_sizes[10] / KDIM;  // 64

    // Workspace layout (all 16B aligned)
    char* ws = (char*)d_ws;
    int*   slot   = (int*)ws;                                  // N * 4 bytes
    size_t off    = (size_t)N * sizeof(int);
    float* Qtab   = (float*)(ws + off);                        // V * 128 * 4
    off          += (size_t)V * E_DIM * sizeof(float);
    float* Ptab   = (float*)(ws + off);                        // 4C * 128 * 4
    off          += (size_t)4 * C * E_DIM * sizeof(float);
    float* tabcat = (float*)(ws + off);                        // 4C * 128 * 4
    (void)ws_size; (void)n_in; (void)out_size;

    // 1) slot map
    init_slot_kernel<<<(N + 255) / 256, 256, 0, stream>>>(slot, N);
    scatter_slot_kernel<<<(4 * M + 255) / 256, 256, 0, stream>>>(
        idx[0], val[0], idx[1], val[1], idx[2], val[2], idx[3], val[3], slot, M);

    // 2) make tabs contiguous [4C, 128]
    for (int k = 0; k < 4; ++k)
        hipMemcpyAsync(tabcat + (size_t)k * C * KDIM, tabs[k],
                       (size_t)C * KDIM * sizeof(float),
                       hipMemcpyDeviceToDevice, stream);

    // 3) WMMA precompute GEMMs (tiny: 1 MB + 128 KB outputs, L2-resident)
    {   // Qtab = fixed_table @ W_fixed + b   : (V/16)*8 tiles, 4 waves/block
        int tiles = (V / 16) * (E_DIM / 16);
        wmma_gemm_f32_kernel<<<(tiles + 3) / 4, 128, 0, stream>>>(ftab, Wf, bias, Qtab, V);
    }
    {   // Ptab = tabcat @ W_sparse           : (4C/16)*8 tiles
        int tiles = (4 * C / 16) * (E_DIM / 16);
        wmma_gemm_f32_kernel<<<(tiles + 3) / 4, 128, 0, stream>>>(tabcat, Ws, nullptr, Ptab, 4 * C);
    }

    // 4) hot gather-add: 8 rows per 256-thread block (1 row per wave32)
    gather_add_kernel<<<(N + 7) / 8, 256, 0, stream>>>(
        ff, slot, (const v4f*)Qtab, (const v4f*)Ptab, (v4f*)d_out, N);
}